// LDDMM_67757404062368
// MI455X (gfx1250) — compile-verified
//
#include <hip/hip_runtime.h>
#include <hip/hip_bf16.h>

#define HH 448
#define WW 448
#define HWp (448*448)          // 200704
#define TT 8
#define NPAIR 16
#define BAND 160               // K(i,j) underflows fp32 beyond |i-j| ~ 133
#define NBLK 784               // HWp / 256
#define NBLK16 12544           // 16*HWp / 256
#define KB 32                  // K-block staged in LDS
#define ALDS_STRIDE 36         // floats; conflict-free b64 A-frag reads
#define BLDS_STRIDE 80         // floats; lane halves hit disjoint bank sets

typedef __attribute__((ext_vector_type(2))) float v2f;
typedef __attribute__((ext_vector_type(4))) float v4f;
typedef __attribute__((ext_vector_type(8))) float v8f;
typedef int b128_t __attribute__((vector_size(4 * sizeof(int))));

// ---------------------------------------------------------------- WMMA core
__device__ inline v8f wmma_f32_16x16x4(v2f a, v2f b, v8f c) {
#if defined(__gfx1250__) && __has_builtin(__builtin_amdgcn_wmma_f32_16x16x4_f32)
  // 8 args: (neg_a, A, neg_b, B, c_mod, C, reuse_a, reuse_b)
  return __builtin_amdgcn_wmma_f32_16x16x4_f32(false, a, false, b, (short)0, c,
                                               false, false);
#else
  // host-pass / fallback placeholder (never used on gfx1250 device)
  c[0] += a.x * b.x + a.y * b.y;
  return c;
#endif
}

// -------------------------------------------------- async global -> LDS copy
#if defined(__gfx1250__) && \
    __has_builtin(__builtin_amdgcn_global_load_async_to_lds_b128) && \
    __has_builtin(__builtin_amdgcn_s_wait_asynccnt)
#define USE_ASYNC_LDS 1
#else
#define USE_ASYNC_LDS 0
#endif

__device__ inline void cp16_async(const float* g, float* l) {
#if USE_ASYNC_LDS
  // param0: b128 __device__* (global src); param1: LDS dst; (offset, cpol)
  float* gnc = const_cast<float*>(g);
  __builtin_amdgcn_global_load_async_to_lds_b128(
      (__attribute__((address_space(1))) b128_t*)gnc,
      (__attribute__((address_space(3))) b128_t*)l, 0, 0);
#else
  *(v4f*)l = *(const v4f*)g;
#endif
}

__device__ inline void async_join() {
#if USE_ASYNC_LDS
  __builtin_amdgcn_s_wait_asynccnt(0);
#endif
  __syncthreads();
}

// ---------------------------------------------------------------- reductions
__device__ inline float block_reduce_sum(float v, float* sm) {
  int tid = threadIdx.x;
  sm[tid] = v;
  __syncthreads();
  for (int s = 128; s > 0; s >>= 1) {
    if (tid < s) sm[tid] += sm[tid + s];
    __syncthreads();
  }
  float r = sm[0];
  __syncthreads();
  return r;
}

// ---------------------------------------------------------------- smoothing
// 11-tap Gaussian (sigma=1.5, r=5), zero padding. 16 fields (T*2).
__global__ __launch_bounds__(256) void k_conv_v(const float* __restrict__ in,
                                                float* __restrict__ out) {
  int gid = blockIdx.x * 256 + threadIdx.x;          // 16*HWp threads
  int f = gid / HWp; int r = gid - f * HWp;
  int y = r / WW;    int x = r - y * WW;
  float gsum = 0.f, acc = 0.f;
  const float* base = in + (size_t)f * HWp;
#pragma unroll
  for (int i = 0; i < 11; ++i) {
    float d = (float)(i - 5);
    float g = __expf(-d * d * (1.0f / 4.5f));
    gsum += g;
    int yy = y + i - 5;
    if (yy >= 0 && yy < HH) acc += g * base[yy * WW + x];
  }
  out[gid] = acc / gsum;
}

__global__ __launch_bounds__(256) void k_conv_h(const float* __restrict__ in,
                                                float* __restrict__ out) {
  int gid = blockIdx.x * 256 + threadIdx.x;
  int f = gid / HWp; int r = gid - f * HWp;
  int y = r / WW;    int x = r - y * WW;
  float gsum = 0.f, acc = 0.f;
  const float* base = in + (size_t)f * HWp + y * WW;
#pragma unroll
  for (int i = 0; i < 11; ++i) {
    float d = (float)(i - 5);
    float g = __expf(-d * d * (1.0f / 4.5f));
    gsum += g;
    int xx = x + i - 5;
    if (xx >= 0 && xx < WW) acc += g * base[xx];
  }
  out[gid] = acc / gsum;
}

// ---------------------------------------------------------------- sampling
__device__ inline float bilin1(const float* __restrict__ img, float xf, float yf) {
  xf = fminf(fmaxf(xf, 0.f), (float)(WW - 1));
  yf = fminf(fmaxf(yf, 0.f), (float)(HH - 1));
  int x0 = (int)floorf(xf), y0 = (int)floorf(yf);
  int x1 = min(x0 + 1, WW - 1), y1 = min(y0 + 1, HH - 1);
  float wx = xf - (float)x0, wy = yf - (float)y0;
  float w00 = (1.f - wx) * (1.f - wy), w01 = wx * (1.f - wy);
  float w10 = (1.f - wx) * wy,         w11 = wx * wy;
  return img[y0 * WW + x0] * w00 + img[y0 * WW + x1] * w01 +
         img[y1 * WW + x0] * w10 + img[y1 * WW + x1] * w11;
}

// flow is identical across the batch (velocity is broadcast, flow starts at 0)
__global__ __launch_bounds__(256) void k_flow(const float* __restrict__ vs,
                                              float* __restrict__ flow) {
  int idx = blockIdx.x * 256 + threadIdx.x;          // HWp threads
  int y = idx / WW, x = idx - y * WW;
  float fx = 0.f, fy = 0.f;
  const float dt = 1.0f / (float)TT;
  for (int t = 0; t < TT; ++t) {
    const float* vt = vs + (size_t)t * 2 * HWp;
    float sx = (float)x + fx, sy = (float)y + fy;
    float vx = bilin1(vt, sx, sy);
    float vy = bilin1(vt + HWp, sx, sy);
    fx += dt * vx;
    fy += dt * vy;
  }
  flow[idx] = fx;
  flow[HWp + idx] = fy;
}

// warp source, write raw warped output, block-partial sums of relu(warped)
__global__ __launch_bounds__(256) void k_warp(const float* __restrict__ src,
                                              const float* __restrict__ flow,
                                              float* __restrict__ warped_out,
                                              float* __restrict__ wpart) {
  __shared__ float sm[256];
  int b = blockIdx.y;
  int idx = blockIdx.x * 256 + threadIdx.x;
  int y = idx / WW, x = idx - y * WW;
  float val = bilin1(src + (size_t)b * HWp, (float)x + flow[idx],
                     (float)y + flow[HWp + idx]);
  warped_out[(size_t)b * HWp + idx] = val;
  float s = block_reduce_sum(fmaxf(val, 0.f), sm);
  if (threadIdx.x == 0) wpart[b * NBLK + blockIdx.x] = s;
}

// replicate flow into d_out for each batch element
__global__ __launch_bounds__(256) void k_flowout(const float* __restrict__ flow,
                                                 float* __restrict__ out_flow) {
  int gid = blockIdx.x * 256 + threadIdx.x;          // 8*HWp threads
  out_flow[gid] = flow[gid % (2 * HWp)];
}

// ---------------------------------------------------------------- reductions
__global__ __launch_bounds__(256) void k_npart(const float* __restrict__ nb,
                                               float* __restrict__ npart) {
  __shared__ float sm[256];
  int n = blockIdx.y;
  int idx = blockIdx.x * 256 + threadIdx.x;
  float s = block_reduce_sum(fmaxf(nb[(size_t)n * HWp + idx], 0.f), sm);
  if (threadIdx.x == 0) npart[n * NBLK + blockIdx.x] = s;
}

__global__ __launch_bounds__(256) void k_gradpart(const float* __restrict__ vel,
                                                  float* __restrict__ gxp,
                                                  float* __restrict__ gyp) {
  __shared__ float sm[256];
  int gid = blockIdx.x * 256 + threadIdx.x;          // 16*HWp threads
  int f = gid / HWp; int r = gid - f * HWp;
  int y = r / WW;    int x = r - y * WW;
  const float* base = vel + (size_t)f * HWp;
  float v0 = base[r];
  float gx = (x < WW - 1) ? (base[r + 1] - v0) : 0.f;
  float gy = (y < HH - 1) ? (base[r + WW] - v0) : 0.f;
  float sx = block_reduce_sum(gx * gx, sm);
  float sy = block_reduce_sum(gy * gy, sm);
  if (threadIdx.x == 0) {
    gxp[blockIdx.x] = sx;
    gyp[blockIdx.x] = sy;
  }
}

// deterministic final sums of the 8 normalization totals
__global__ void k_sums8(const float* __restrict__ wpart,
                        const float* __restrict__ npart,
                        float* __restrict__ sums) {
  int i = threadIdx.x;                                // 8 threads
  const float* p = (i < 4) ? (wpart + i * NBLK) : (npart + (i - 4) * NBLK);
  float s = 0.f;
  for (int k = 0; k < NBLK; ++k) s += p[k];
  sums[i] = s;
}

__global__ __launch_bounds__(256) void k_build_mu_nu(const float* __restrict__ warped,
                                                     const float* __restrict__ nb,
                                                     const float* __restrict__ sums,
                                                     float* __restrict__ mu,
                                                     float* __restrict__ nu) {
  int b = blockIdx.y;
  int idx = blockIdx.x * 256 + threadIdx.x;
  mu[(size_t)b * HWp + idx] = fmaxf(warped[(size_t)b * HWp + idx], 0.f) / (sums[b] + 1e-12f);
  nu[(size_t)b * HWp + idx] = fmaxf(nb[(size_t)b * HWp + idx], 0.f) / (sums[4 + b] + 1e-12f);
}

__global__ __launch_bounds__(256) void k_fill1(float* __restrict__ dst) {
  dst[blockIdx.x * 256 + threadIdx.x] = 1.0f;
}

// K = exp(-((i-j)/447)^2 / eps), KC = K * C (H==W so KH==KW, KH*CH==KW*CW)
__global__ __launch_bounds__(256) void k_buildK(float* __restrict__ K,
                                                float* __restrict__ KC) {
  int idx = blockIdx.x * 256 + threadIdx.x;          // HWp threads
  int i = idx / WW, j = idx - i * WW;
  float d = (float)(i - j) * (1.0f / 447.0f);
  float c = d * d;
  float k = __expf(-c * 1000.0f);                    // eps = 0.001
  K[idx] = k;
  KC[idx] = k * c;
}

// ---------------------------------------------------------------- WMMA GEMM
// mode 0: Y[p] = G * X[p];  mode 1: Y[p] = X[p] * G.   All 448x448 row-major.
// Block = 256 threads (8 waves), 64x64 output tile; each wave: two 16x16 tiles.
// K-blocks of 32 are staged into LDS with GLOBAL_LOAD_ASYNC_TO_LDS_B128
// (ASYNCcnt), then WMMA fragments come from conflict-free ds_load ops.
// K-loop restricted to the +-BAND diagonal band of G (fp32-exact: K underflows;
// tiles are 64-aligned so the band edges stay KB-aligned).
__global__ __launch_bounds__(256) void k_gemm(const float* __restrict__ G,
                                              const float* __restrict__ X,
                                              float* __restrict__ Y, int mode) {
  __shared__ float Al[64 * ALDS_STRIDE];   //  9216 B
  __shared__ float Bl[KB * BLDS_STRIDE];   // 10240 B

  int p = blockIdx.z;
  const float* Xp = X + (size_t)p * HWp;
  float* Yp = Y + (size_t)p * HWp;

  int tile_n0 = blockIdx.x * 64;
  int tile_m0 = blockIdx.y * 64;

  int tid = threadIdx.x;
  int wave = tid >> 5;                                // 0..7
  int lane = tid & 31;
  int m0w = (wave & 3) * 16;                          // wave row strip in tile
  int tn0w = (wave >> 2) * 32;                        // wave col pair in tile

  // band of k contributing to this 64-wide block of G's banded dimension
  int center = (mode == 0) ? tile_m0 : tile_n0;
  int k_lo = max(center - BAND, 0) & ~(KB - 1);
  int k_hi = min(center + 64 + BAND, 448);

  const float* Amat = (mode == 0) ? G : Xp;           // supplies 16x4 A frags
  const float* Bmat = (mode == 0) ? Xp : G;           // supplies 4x16 B frags

  // staging assignment (two 16B chunks each from A and B per thread):
  // A: 64 rows x 32 floats = 512 chunks; chunk c: row=c>>3, sub=c&7
  // B: KB rows x 64 floats = 512 chunks; chunk c: row=c>>4, sub=c&15
  int ac0r = tid >> 3,  ac0s = (tid & 7) * 4;
  int ac1r = ac0r + 32;                               // chunk tid+256
  int bc0r = tid >> 4,  bc0s = (tid & 15) * 4;
  int bc1r = bc0r + 16;                               // chunk tid+256

  int arow = lane & 15;                               // M (A) / N (B) index
  int koff = (lane >> 4) * 2;                         // K pair select

  v8f acc0 = {};
  v8f acc1 = {};

  for (int kb = k_lo; kb < k_hi; kb += KB) {
    __syncthreads();                                  // LDS reuse fence
    cp16_async(Amat + (size_t)(tile_m0 + ac0r) * WW + kb + ac0s,
               Al + ac0r * ALDS_STRIDE + ac0s);
    cp16_async(Amat + (size_t)(tile_m0 + ac1r) * WW + kb + ac0s,
               Al + ac1r * ALDS_STRIDE + ac0s);
    cp16_async(Bmat + (size_t)(kb + bc0r) * WW + tile_n0 + bc0s,
               Bl + bc0r * BLDS_STRIDE + bc0s);
    cp16_async(Bmat + (size_t)(kb + bc1r) * WW + tile_n0 + bc0s,
               Bl + bc1r * BLDS_STRIDE + bc0s);
    if (kb + KB < k_hi) {                             // uniform-branch prefetch
      __builtin_prefetch(Amat + (size_t)(tile_m0 + ac0r) * WW + kb + KB + ac0s, 0, 0);
      __builtin_prefetch(Bmat + (size_t)(kb + KB + bc0r) * WW + tile_n0 + bc0s, 0, 0);
    }
    async_join();                                     // s_wait_asynccnt + barrier

#pragma unroll
    for (int ks = 0; ks < KB; ks += 4) {
      // A 16x4: VGPR0 = K{0,2}, VGPR1 = K{1,3} across lane halves (b64 read)
      v2f a = *(const v2f*)(Al + (m0w + arow) * ALDS_STRIDE + ks + koff);
      // B 4x16 tiles at cols tn0w and tn0w+16
      const float* bp = Bl + (ks + koff) * BLDS_STRIDE + tn0w + arow;
      v2f b0; b0.x = bp[0];  b0.y = bp[BLDS_STRIDE];
      v2f b1; b1.x = bp[16]; b1.y = bp[BLDS_STRIDE + 16];
      acc0 = wmma_f32_16x16x4(a, b0, acc0);
      acc1 = wmma_f32_16x16x4(a, b1, acc1);
    }
  }

  // C/D layout: VGPR r -> M = m0 + r + (lane>=16 ? 8 : 0), N = n0 + (lane&15)
  int cm = tile_m0 + m0w + (lane >> 4) * 8;
  int cn = tile_n0 + tn0w + (lane & 15);
#pragma unroll
  for (int r = 0; r < 8; ++r) {
    Yp[(size_t)(cm + r) * WW + cn] = acc0[r];
    Yp[(size_t)(cm + r) * WW + cn + 16] = acc1[r];
  }
}

// mode 0: dst=u, numerator mu[b] (b = p>>2); mode 1: dst=v, numerator nu[n] (n = p&3)
__global__ __launch_bounds__(256) void k_div(float* __restrict__ dst,
                                             const float* __restrict__ num,
                                             const float* __restrict__ den,
                                             int mode) {
  int p = blockIdx.y;
  int idx = blockIdx.x * 256 + threadIdx.x;
  int sel = (mode == 0) ? (p >> 2) : (p & 3);
  dst[(size_t)p * HWp + idx] =
      num[(size_t)sel * HWp + idx] / (den[(size_t)p * HWp + idx] + 1e-30f);
}

__global__ __launch_bounds__(256) void k_costpart(const float* __restrict__ u,
                                                  const float* __restrict__ c,
                                                  float* __restrict__ costp,
                                                  int accumulate) {
  __shared__ float sm[256];
  int p = blockIdx.y;
  int idx = blockIdx.x * 256 + threadIdx.x;
  float s = block_reduce_sum(u[(size_t)p * HWp + idx] * c[(size_t)p * HWp + idx], sm);
  if (threadIdx.x == 0) {
    float* dst = costp + p * NBLK + blockIdx.x;
    *dst = accumulate ? (*dst + s) : s;
  }
}

__global__ void k_finalize(const float* __restrict__ weights,
                           const float* __restrict__ costp,
                           const float* __restrict__ gxp,
                           const float* __restrict__ gyp,
                           float* __restrict__ out3) {
  // single thread; deterministic sequential sums
  float wsum = weights[0] + weights[1] + weights[2] + weights[3] + 1e-12f;
  float loss_w2 = 0.f;
  for (int p = 0; p < NPAIR; ++p) {
    float c = 0.f;
    for (int k = 0; k < NBLK; ++k) c += costp[p * NBLK + k];
    loss_w2 += c * (weights[p & 3] / wsum);           // n = p % 4
  }
  float sgx = 0.f, sgy = 0.f;
  for (int k = 0; k < NBLK16; ++k) { sgx += gxp[k]; sgy += gyp[k]; }
  const float ngx = (float)TT * 2.f * HH * (WW - 1);
  const float ngy = (float)TT * 2.f * (HH - 1) * WW;
  float loss_reg = (sgx / ngx + sgy / ngy) * 0.001f;
  out3[0] = loss_w2 + loss_reg;
  out3[1] = loss_w2;
  out3[2] = loss_reg;
}

// ---------------------------------------------------------------- launch
extern "C" void kernel_launch(void* const* d_in, const int* in_sizes, int n_in,
                              void* d_out, int out_size, void* d_ws, size_t ws_size,
                              hipStream_t stream) {
  (void)in_sizes; (void)n_in; (void)out_size; (void)ws_size;
  const float* src = (const float*)d_in[0];   // (4,1,448,448)
  const float* nb  = (const float*)d_in[1];   // (4,448,448)
  const float* wts = (const float*)d_in[2];   // (4,)
  const float* vel = (const float*)d_in[3];   // (8,2,448,448)
  float* out = (float*)d_out;
  float* warped_out = out;                    // 4*HWp
  float* flow_out   = out + (size_t)4 * HWp;  // 8*HWp
  float* out3       = out + (size_t)12 * HWp; // 3 scalars

  float* w = (float*)d_ws;
  float* t1   = w;                         // 16*HWp (also conv temp)
  float* t2   = w + (size_t)16 * HWp;      // 16*HWp (also smoothed velocity)
  float* u    = w + (size_t)32 * HWp;      // 16*HWp
  float* v    = w + (size_t)48 * HWp;      // 16*HWp
  float* mu   = w + (size_t)64 * HWp;      // 4*HWp
  float* nu   = w + (size_t)68 * HWp;      // 4*HWp
  float* flw  = w + (size_t)72 * HWp;      // 2*HWp
  float* Kmat = w + (size_t)74 * HWp;      // HWp
  float* KC   = w + (size_t)75 * HWp;      // HWp
  float* wpart = w + (size_t)76 * HWp;             // 4*NBLK
  float* npart = wpart + 4 * NBLK;                 // 4*NBLK
  float* gxp   = npart + 4 * NBLK;                 // NBLK16
  float* gyp   = gxp + NBLK16;                     // NBLK16
  float* costp = gyp + NBLK16;                     // 16*NBLK
  float* sums  = costp + NPAIR * NBLK;             // 8

  const dim3 B256(256);
  const dim3 gHW(NBLK), gHW4(NBLK, 4), gHW16(NBLK, 16);
  const dim3 g16(NBLK16), gGemm(7, 7, 16);

  // velocity smoothing (separable Gaussian) -> t2
  k_conv_v<<<g16, B256, 0, stream>>>(vel, t1);
  k_conv_h<<<g16, B256, 0, stream>>>(t1, t2);

  // flow integration (batch-invariant) + warp + flow replication
  k_flow<<<gHW, B256, 0, stream>>>(t2, flw);
  k_warp<<<gHW4, B256, 0, stream>>>(src, flw, warped_out, wpart);
  k_flowout<<<dim3(8 * HWp / 256), B256, 0, stream>>>(flw, flow_out);

  // normalization + regularizer reductions
  k_npart<<<gHW4, B256, 0, stream>>>(nb, npart);
  k_gradpart<<<g16, B256, 0, stream>>>(vel, gxp, gyp);
  k_sums8<<<1, 8, 0, stream>>>(wpart, npart, sums);
  k_build_mu_nu<<<gHW4, B256, 0, stream>>>(warped_out, nb, sums, mu, nu);

  // Sinkhorn
  k_fill1<<<g16, B256, 0, stream>>>(v);
  k_buildK<<<gHW, B256, 0, stream>>>(Kmat, KC);
  for (int it = 0; it < 5; ++it) {
    k_gemm<<<gGemm, B256, 0, stream>>>(Kmat, v, t1, 0);   // t1 = K*v
    k_gemm<<<gGemm, B256, 0, stream>>>(Kmat, t1, t2, 1);  // t2 = t1*K
    k_div<<<gHW16, B256, 0, stream>>>(u, mu, t2, 0);      // u = mu/(t2+TINY)
    k_gemm<<<gGemm, B256, 0, stream>>>(Kmat, u, t1, 0);
    k_gemm<<<gGemm, B256, 0, stream>>>(Kmat, t1, t2, 1);
    k_div<<<gHW16, B256, 0, stream>>>(v, nu, t2, 1);      // v = nu/(t2+TINY)
  }
  // cost = sum u * (KC*v*K + K*v*KC)
  k_gemm<<<gGemm, B256, 0, stream>>>(KC, v, t1, 0);
  k_gemm<<<gGemm, B256, 0, stream>>>(Kmat, t1, t2, 1);
  k_costpart<<<gHW16, B256, 0, stream>>>(u, t2, costp, 0);
  k_gemm<<<gGemm, B256, 0, stream>>>(Kmat, v, t1, 0);
  k_gemm<<<gGemm, B256, 0, stream>>>(KC, t1, t2, 1);
  k_costpart<<<gHW16, B256, 0, stream>>>(u, t2, costp, 1);

  k_finalize<<<1, 1, 0, stream>>>(wts, costp, gxp, gyp, out3);
}